// POLYCORE_41008347742525
// MI455X (gfx1250) — compile-verified
//
#include <hip/hip_runtime.h>
#include <math.h>

// CDNA5 / gfx1250 fused contour-refinement kernel.
// One workgroup (256 thr = 8 wave32) per batch sample; all intermediate
// state lives in LDS (CDNA5: up to 320KB per workgroup). GEMMs use
// V_WMMA_F32_16X16X4_F32 (fp32, matches reference precision).
// K-loops unrolled x8 so many B-operand load groups are in flight
// between dependent WMMAs (latency-bound workload).

typedef float v2f __attribute__((ext_vector_type(2)));
typedef float v8f __attribute__((ext_vector_type(8)));

#define GD 128
#define INP_STRIDE 648   // 644 cols (max din padded to mult of 4) + 4 pad

__device__ __forceinline__ int ring_prev(int n, int Ph) {
  int c = (n >= Ph) ? 1 : 0; int j = n - c * Ph;
  return c * Ph + ((j == 0) ? (Ph - 1) : (j - 1));
}
__device__ __forceinline__ int ring_next(int n, int Ph) {
  int c = (n >= Ph) ? 1 : 0; int j = n - c * Ph;
  return c * Ph + ((j == Ph - 1) ? 0 : (j + 1));
}
__device__ __forceinline__ float lane_sum32(float v) {
#pragma unroll
  for (int off = 16; off > 0; off >>= 1) v += __shfl_xor(v, off, 32);
  return v;
}

__global__ void __launch_bounds__(256, 1)
contour_refine_kernel(const float* __restrict__ f0, const float* __restrict__ f1,
                      const float* __restrict__ f2, const float* __restrict__ f3,
                      const float* __restrict__ points,
                      const float* __restrict__ w1_1, const float* __restrict__ b1_1,
                      const float* __restrict__ wr_1, const float* __restrict__ br_1,
                      const float* __restrict__ w1_r, const float* __restrict__ b1_r,
                      const float* __restrict__ wr_r, const float* __restrict__ br_r,
                      const float* __restrict__ w2a, const float* __restrict__ b2a,
                      const float* __restrict__ woa, const float* __restrict__ boa,
                      float* __restrict__ out, int bs)
{
  extern __shared__ float smem[];
  float* bufA = smem;                    // 128*128  (x / h)
  float* bufB = bufA + 128 * GD;         // 128*128  (h1 / upsample target)
  float* Tb   = bufB + 128 * GD;         // 128*128  (t1 / t2)
  float* Rb   = Tb   + 128 * GD;         // 128*128  (residual inp@Wr)
  float* INP  = Rb   + 128 * GD;         // 16*INP_STRIDE staging tile
  float* PA   = INP  + 16 * INP_STRIDE;  // 128*2 points
  float* PB   = PA + 256;
  float* NRM  = PB + 256;                // 128*2 normals
  float* RAW  = NRM + 256;               // 128*2 head pre-smooth
  float* s_b1 = RAW + 256;               // 128 level-constant staging
  float* s_b2 = s_b1 + GD;               // 128
  float* s_br = s_b2 + GD;               // 128
  float* s_wo = s_br + GD;               // 256
  float* s_wx = s_wo + 256;              // 64 sampling params
  float* s_wy = s_wx + 64;
  int*   s_x0 = (int*)(s_wy + 64);
  int*   s_x1 = s_x0 + 64;
  int*   s_y0 = s_x1 + 64;
  int*   s_y1 = s_y0 + 64;

  const int bat  = blockIdx.x;
  const int tid  = threadIdx.x;
  const int lane = tid & 31;
  const int wave = tid >> 5;
  const int lm   = lane & 15;   // WMMA: row / col within tile
  const int hi   = lane >> 4;   // WMMA: K-pair / M-half selector

  float* xcur = bufA;
  float* xoth = bufB;
  float* pcur = PA;
  float* palt = PB;

  if (tid < 32) pcur[tid] = points[bat * 32 + tid];
  __syncthreads();

  int N = 16;
  for (int lvl = 0; lvl < 4; ++lvl) {
    const int Ph    = N >> 1;
    const int din   = (lvl == 0) ? 514 : 642;
    const int foff  = (lvl == 0) ? 0 : GD;
    const int kmain = din & ~3;          // 512 / 640 (tail of 2 valid cols)
    const float* W1p = (lvl == 0) ? w1_1 : (w1_r + (size_t)(lvl - 1) * 642 * GD);
    const float* Wrp = (lvl == 0) ? wr_1 : (wr_r + (size_t)(lvl - 1) * 642 * GD);
    const float* b1p = (lvl == 0) ? b1_1 : (b1_r + (lvl - 1) * GD);
    const float* brp = (lvl == 0) ? br_1 : (br_r + (lvl - 1) * GD);
    const float* W2p = w2a + (size_t)lvl * GD * GD;
    const float* b2p = b2a + lvl * GD;
    const float* wop = woa + lvl * GD * 2;
    const float* bop = boa + lvl * 2;
    const int ooff   = ((lvl == 0) ? 0 : (lvl == 1) ? 32 : (lvl == 2) ? 96 : 224) * bs;

    // ---- stage small per-level constants in LDS ----
    if (tid < GD) {
      s_b1[tid] = b1p[tid];
      s_b2[tid] = b2p[tid];
      s_br[tid] = brp[tid];
    }
    s_wo[tid] = wop[tid];
    // ---- S0: ring normals from current points ----
    if (tid < N) {
      int n = tid;
      int pm = ring_prev(n, Ph), np = ring_next(n, Ph);
      float px = pcur[n * 2], py = pcur[n * 2 + 1];
      float ax = pcur[pm * 2] - px, ay = pcur[pm * 2 + 1] - py;
      float il = rsqrtf(ax * ax + ay * ay);
      float n1x = -ay * il, n1y = ax * il;
      float bx = px - pcur[np * 2], by = py - pcur[np * 2 + 1];
      il = rsqrtf(bx * bx + by * by);
      float n2x = -by * il, n2y = bx * il;
      float sx = n1x + n2x, sy = n1y + n2y;
      il = rsqrtf(sx * sx + sy * sy);
      NRM[n * 2] = sx * il; NRM[n * 2 + 1] = sy * il;
    }
    __syncthreads();

    // ---- S1: Tb = inp@W1, Rb = inp@Wr  (M-tiled; sample f on the fly) ----
    const int Mt = N >> 4;
    for (int mt = 0; mt < Mt; ++mt) {
      const int mbase = mt << 4;
      // bilinear sampling parameters per (node-in-tile, pyramid level)
      if (tid < 64) {
        int i = tid >> 2, map = tid & 3;
        int n = mbase + i;
        int Wm = (map == 0) ? 224 : (map == 1) ? 112 : (map == 2) ? 56 : 28;
        float px = pcur[n * 2 + 0], py = pcur[n * 2 + 1];
        float gx = fminf(fmaxf(2.f * px - 1.f, -1.f), 1.f);
        float gy = fminf(fmaxf(2.f * py - 1.f, -1.f), 1.f);
        float ix = (gx + 1.f) * 0.5f * (float)(Wm - 1);
        float iy = (gy + 1.f) * 0.5f * (float)(Wm - 1);
        float x0 = floorf(ix), y0 = floorf(iy);
        s_wx[tid] = ix - x0; s_wy[tid] = iy - y0;
        int x0i = (int)x0, y0i = (int)y0;
        s_x0[tid] = min(max(x0i, 0), Wm - 1);
        s_x1[tid] = min(max(x0i + 1, 0), Wm - 1);
        s_y0[tid] = min(max(y0i, 0), Wm - 1);
        s_y1[tid] = min(max(y0i + 1, 0), Wm - 1);
      }
      __syncthreads();
      // fill staging tile: [x (lvl>0) | sampled f (512) | p | zero pad]
      for (int idx = tid; idx < 16 * 512; idx += 256) {
        int i = idx >> 9, ff = idx & 511;
        int map = ff >> 7, c = ff & 127;
        int si = (i << 2) + map;
        const float* fp = (map == 0) ? f0 : (map == 1) ? f1 : (map == 2) ? f2 : f3;
        int Wm = (map == 0) ? 224 : (map == 1) ? 112 : (map == 2) ? 56 : 28;
        size_t pb = ((size_t)(bat * GD + c)) * (size_t)(Wm * Wm);
        int x0 = s_x0[si], x1 = s_x1[si], y0 = s_y0[si], y1 = s_y1[si];
        float wx = s_wx[si], wy = s_wy[si];
        float v00 = fp[pb + (size_t)y0 * Wm + x0];
        float v01 = fp[pb + (size_t)y0 * Wm + x1];
        float v10 = fp[pb + (size_t)y1 * Wm + x0];
        float v11 = fp[pb + (size_t)y1 * Wm + x1];
        INP[i * INP_STRIDE + foff + ff] =
            v00 * (1.f - wx) * (1.f - wy) + v01 * wx * (1.f - wy) +
            v10 * (1.f - wx) * wy + v11 * wx * wy;
      }
      if (lvl > 0) {
        for (int idx = tid; idx < 16 * GD; idx += 256) {
          int i = idx >> 7, f = idx & 127;
          INP[i * INP_STRIDE + f] = xcur[(mbase + i) * GD + f];
        }
      }
      if (tid < 64) {  // p columns + 2 zero-pad columns
        int i = tid >> 2, j = tid & 3;
        INP[i * INP_STRIDE + (din - 2) + j] = (j < 2) ? pcur[(mbase + i) * 2 + j] : 0.f;
      }
      __syncthreads();
      // each wave: one 16x16 tile vs W1 and vs Wr (shared A loads)
      {
        const int n0 = wave << 4;
        const int col = n0 + lm;
        v8f acc1 = {0.f, 0.f, 0.f, 0.f, 0.f, 0.f, 0.f, 0.f};
        v8f accr = {0.f, 0.f, 0.f, 0.f, 0.f, 0.f, 0.f, 0.f};
        const float* Ar = INP + (size_t)lm * INP_STRIDE;
#pragma unroll 8
        for (int k0 = 0; k0 < kmain; k0 += 4) {
          const int ka = k0 + (hi << 1);
          v2f a   = { Ar[ka], Ar[ka + 1] };
          v2f b1v = { W1p[(size_t)ka * GD + col], W1p[(size_t)(ka + 1) * GD + col] };
          v2f brv = { Wrp[(size_t)ka * GD + col], Wrp[(size_t)(ka + 1) * GD + col] };
          acc1 = __builtin_amdgcn_wmma_f32_16x16x4_f32(false, a, false, b1v, (short)0, acc1, false, false);
          accr = __builtin_amdgcn_wmma_f32_16x16x4_f32(false, a, false, brv, (short)0, accr, false, false);
        }
        {  // K tail: cols kmain..kmain+1 valid, +2..+3 zero (LDS already zero)
          const int ka = kmain + (hi << 1);
          v2f a   = { Ar[ka], Ar[ka + 1] };
          v2f b1v = { hi ? 0.f : W1p[(size_t)kmain * GD + col],
                      hi ? 0.f : W1p[(size_t)(kmain + 1) * GD + col] };
          v2f brv = { hi ? 0.f : Wrp[(size_t)kmain * GD + col],
                      hi ? 0.f : Wrp[(size_t)(kmain + 1) * GD + col] };
          acc1 = __builtin_amdgcn_wmma_f32_16x16x4_f32(false, a, false, b1v, (short)0, acc1, false, false);
          accr = __builtin_amdgcn_wmma_f32_16x16x4_f32(false, a, false, brv, (short)0, accr, false, false);
        }
#pragma unroll
        for (int r = 0; r < 8; ++r) {
          int row = mbase + r + (hi << 3);
          Tb[(size_t)row * GD + col] = acc1[r];
          Rb[(size_t)row * GD + col] = accr[r];
        }
      }
      __syncthreads();
    }

    // ---- S2: h1_pre = smooth3(Tb) + b1  -> xoth ----
    for (int idx = tid; idx < N * GD; idx += 256) {
      int n = idx >> 7, f = idx & 127;
      int pm = ring_prev(n, Ph), np = ring_next(n, Ph);
      xoth[idx] = (Tb[pm * GD + f] + Tb[idx] + Tb[np * GD + f]) * (1.f / 3.f) + s_b1[f];
    }
    __syncthreads();
    // ---- S3: instance-norm + relu in place (wave per row) ----
    for (int n = wave; n < N; n += 8) {
      float* row = xoth + n * GD;
      float v0 = row[lane], v1 = row[lane + 32], v2 = row[lane + 64], v3 = row[lane + 96];
      float m = lane_sum32(v0 + v1 + v2 + v3) * (1.f / 128.f);
      float d0 = v0 - m, d1 = v1 - m, d2 = v2 - m, d3 = v3 - m;
      float var = lane_sum32(d0 * d0 + d1 * d1 + d2 * d2 + d3 * d3) * (1.f / 128.f);
      float sc = rsqrtf(var + 1e-5f);
      row[lane]      = fmaxf(d0 * sc, 0.f);
      row[lane + 32] = fmaxf(d1 * sc, 0.f);
      row[lane + 64] = fmaxf(d2 * sc, 0.f);
      row[lane + 96] = fmaxf(d3 * sc, 0.f);
    }
    __syncthreads();
    // ---- S4: Tb = h1 @ W2 (K=128) ----
    for (int t = wave; t < Mt * 8; t += 8) {
      const int mb = (t >> 3) << 4;
      const int n0 = (t & 7) << 4;
      const int col = n0 + lm;
      v8f acc = {0.f, 0.f, 0.f, 0.f, 0.f, 0.f, 0.f, 0.f};
      const float* Ar = xoth + (size_t)(mb + lm) * GD;
#pragma unroll 8
      for (int k0 = 0; k0 < GD; k0 += 4) {
        const int ka = k0 + (hi << 1);
        v2f a  = { Ar[ka], Ar[ka + 1] };
        v2f bv = { W2p[(size_t)ka * GD + col], W2p[(size_t)(ka + 1) * GD + col] };
        acc = __builtin_amdgcn_wmma_f32_16x16x4_f32(false, a, false, bv, (short)0, acc, false, false);
      }
#pragma unroll
      for (int r = 0; r < 8; ++r)
        Tb[(size_t)(mb + r + (hi << 3)) * GD + col] = acc[r];
    }
    __syncthreads();
    // ---- S5: h2_pre = smooth3(Tb) + b2 -> xcur (old x dead) ----
    for (int idx = tid; idx < N * GD; idx += 256) {
      int n = idx >> 7, f = idx & 127;
      int pm = ring_prev(n, Ph), np = ring_next(n, Ph);
      xcur[idx] = (Tb[pm * GD + f] + Tb[idx] + Tb[np * GD + f]) * (1.f / 3.f) + s_b2[f];
    }
    __syncthreads();
    // ---- S6: inorm + relu + residual (Rb + br) in place ----
    for (int n = wave; n < N; n += 8) {
      float* row = xcur + n * GD;
      const float* rr = Rb + n * GD;
      float v0 = row[lane], v1 = row[lane + 32], v2 = row[lane + 64], v3 = row[lane + 96];
      float m = lane_sum32(v0 + v1 + v2 + v3) * (1.f / 128.f);
      float d0 = v0 - m, d1 = v1 - m, d2 = v2 - m, d3 = v3 - m;
      float var = lane_sum32(d0 * d0 + d1 * d1 + d2 * d2 + d3 * d3) * (1.f / 128.f);
      float sc = rsqrtf(var + 1e-5f);
      row[lane]      = fmaxf(d0 * sc, 0.f) + rr[lane]      + s_br[lane];
      row[lane + 32] = fmaxf(d1 * sc, 0.f) + rr[lane + 32] + s_br[lane + 32];
      row[lane + 64] = fmaxf(d2 * sc, 0.f) + rr[lane + 64] + s_br[lane + 64];
      row[lane + 96] = fmaxf(d3 * sc, 0.f) + rr[lane + 96] + s_br[lane + 96];
    }
    __syncthreads();
    // ---- S8: magnitude head, RAW = h @ wo (wo staged in LDS) ----
    if (tid < N * 2) {
      int n = tid >> 1, j = tid & 1;
      const float* row = xcur + n * GD;
      float s = 0.f;
#pragma unroll 8
      for (int f = 0; f < GD; ++f) s += row[f] * s_wo[f * 2 + j];
      RAW[tid] = s;
    }
    __syncthreads();
    // ---- S9: mag = sigmoid(smooth3(RAW)+bo)-0.5 ; p += STEP*n*mag ; emit ----
    if (tid < N * 2) {
      int n = tid >> 1, j = tid & 1;
      int pm = ring_prev(n, Ph), np = ring_next(n, Ph);
      float mo = (RAW[pm * 2 + j] + RAW[tid] + RAW[np * 2 + j]) * (1.f / 3.f) + bop[j];
      float mag = 1.f / (1.f + expf(-mo)) - 0.5f;
      float pn = pcur[tid] + 0.05f * NRM[tid] * mag;
      palt[tid] = pn;
      out[ooff + bat * (N * 2) + tid] = pn;
    }
    __syncthreads();
    // ---- S10: 2x contour upsample of h and p ----
    if (lvl < 3) {
      for (int idx = tid; idx < 2 * N * GD; idx += 256) {
        int o = idx >> 7, f = idx & 127;
        int c = (o >= N) ? 1 : 0;
        int jo = o - c * N;
        int j = jo >> 1;
        int s0 = c * Ph + j;
        float v = xcur[s0 * GD + f];
        if (jo & 1) {
          int s1 = c * Ph + ((j + 1 == Ph) ? 0 : j + 1);
          v = 0.5f * (v + xcur[s1 * GD + f]);
        }
        xoth[o * GD + f] = v;
      }
      if (tid < 4 * N) {
        int o = tid >> 1, j = tid & 1;
        int c = (o >= N) ? 1 : 0;
        int jo = o - c * N;
        int jj = jo >> 1;
        int s0 = c * Ph + jj;
        float v = palt[s0 * 2 + j];
        if (jo & 1) {
          int s1 = c * Ph + ((jj + 1 == Ph) ? 0 : jj + 1);
          v = 0.5f * (v + palt[s1 * 2 + j]);
        }
        pcur[o * 2 + j] = v;
      }
      float* tmp = xcur; xcur = xoth; xoth = tmp;
      N <<= 1;
      __syncthreads();
    }
  }
}

extern "C" void kernel_launch(void* const* d_in, const int* in_sizes, int n_in,
                              void* d_out, int out_size, void* d_ws, size_t ws_size,
                              hipStream_t stream) {
  (void)n_in; (void)out_size; (void)d_ws; (void)ws_size;
  const float* f0     = (const float*)d_in[0];
  const float* f1     = (const float*)d_in[1];
  const float* f2     = (const float*)d_in[2];
  const float* f3     = (const float*)d_in[3];
  const float* points = (const float*)d_in[4];
  const float* w1_1   = (const float*)d_in[8];
  const float* b1_1   = (const float*)d_in[9];
  const float* wr_1   = (const float*)d_in[10];
  const float* br_1   = (const float*)d_in[11];
  const float* w1_r   = (const float*)d_in[12];
  const float* b1_r   = (const float*)d_in[13];
  const float* wr_r   = (const float*)d_in[14];
  const float* br_r   = (const float*)d_in[15];
  const float* w2     = (const float*)d_in[16];
  const float* b2     = (const float*)d_in[17];
  const float* wo     = (const float*)d_in[18];
  const float* bo     = (const float*)d_in[19];
  float* out = (float*)d_out;

  const int bs = in_sizes[4] / 32;  // points: (bs, 2, 8, 2)
  const size_t smem_bytes =
      (size_t)(4 * 128 * GD + 16 * INP_STRIDE + 4 * 256 + 640 + 6 * 64) * sizeof(float);
  hipFuncSetAttribute(reinterpret_cast<const void*>(contour_refine_kernel),
                      hipFuncAttributeMaxDynamicSharedMemorySize, (int)smem_bytes);
  contour_refine_kernel<<<bs, 256, smem_bytes, stream>>>(
      f0, f1, f2, f3, points, w1_1, b1_1, wr_1, br_1,
      w1_r, b1_r, wr_r, br_r, w2, b2, wo, bo, out, bs);
}